// MultiScalePatchEmbedding_23046794510867
// MI455X (gfx1250) — compile-verified
//
#include <hip/hip_runtime.h>
#include <hip/hip_bf16.h>

// MultiScalePatchEmbedding, algebraically collapsed:
//   out[b,m,:] = Acat[b*64+m, 0:56] @ Wtot[0:56,:] + b_total
// where Acat is a per-scale coefficient-weighted reduction of x over tokens
// and Wtot = (proj_w_s @ fuse_block_s @ flat_w) stacked over scales.
// Dominant cost: one 16 MiB read of x (~0.7us @ 23.3 TB/s). Final GEMM uses
// exact-fp32 V_WMMA_F32_16X16X4_F32.

#define B_ 16
#define L_ 4096
#define M_ 64
#define SPLITS 8          // token-splits for deterministic two-stage reduction

typedef __attribute__((ext_vector_type(2))) float v2f;
typedef __attribute__((ext_vector_type(8))) float v8f;

// ---- resample+mean coefficients per scale: c_i = sum_t weight(i,t)/512 ----
__global__ void k_coef(float* c0, float* c1, float* c2) {
    int t = threadIdx.x;                       // blockDim = 512
    const int Ns[3] = {512, 256, 128};
    float* cs[3] = {c0, c1, c2};
    for (int sc = 0; sc < 3; ++sc) {
        int N = Ns[sc];
        if (t < N) {
            double acc = 0.0;
            double step = (double)(N - 1) / 511.0;
            for (int tt = 0; tt < 512; ++tt) {
                double pos = tt * step;
                int i0 = (int)pos;             // pos >= 0
                int i1 = i0 + 1; if (i1 > N - 1) i1 = N - 1;
                double w1 = pos - (double)i0;
                if (t == i0) acc += 1.0 - w1;
                if (t == i1) acc += w1;
            }
            cs[sc][t] = (float)(acc / 512.0);
        }
    }
}

// ---- stage 1: one coalesced pass over x, all 3 scales per loaded element ---
// e2 indexes within a scale-2 tile (2048 floats, contiguous). For fixed e2 the
// scale-1 half (h) and scale-0 quarter (q) are constants -> uniform c loads.
__global__ void k_stage1(const float* __restrict__ x,
                         const float* __restrict__ c0,
                         const float* __restrict__ c1,
                         const float* __restrict__ c2,
                         float* __restrict__ p0, float* __restrict__ p1,
                         float* __restrict__ p2) {
    int e2 = blockIdx.x * 256 + threadIdx.x;   // 0..2047
    int b  = blockIdx.y;                       // 0..15
    int sp = blockIdx.z;                       // 0..SPLITS-1
    int h  = e2 >> 10;                         // 0..1
    int q  = e2 >> 9;                          // 0..3
    const float* xb = x + (size_t)b * (L_ * M_);
    int i2b = sp * (128 / SPLITS);
    float a0 = 0.f, a1 = 0.f, a2 = 0.f;
#pragma unroll 4
    for (int i = 0; i < 128 / SPLITS; ++i) {
        int i2 = i2b + i;
        float v = xb[(size_t)i2 * 2048 + e2];
        a2 = fmaf(c2[i2],         v, a2);
        a1 = fmaf(c1[2 * i2 + h], v, a1);
        a0 = fmaf(c0[4 * i2 + q], v, a0);
    }
    size_t o = ((size_t)sp * B_ + b) * 2048 + e2;
    p0[o] = a0; p1[o] = a1; p2[o] = a2;
}

// ---- stage 2: combine splits/sub-tiles into Acat[1024][64] (cols 56..63=0) -
__global__ void k_stage2(const float* __restrict__ p0,
                         const float* __restrict__ p1,
                         const float* __restrict__ p2,
                         float* __restrict__ A) {
    int idx = blockIdx.x * 256 + threadIdx.x;  // 0..65535
    int row = idx >> 6;                        // b*64 + m
    int kk  = idx & 63;
    int b = row >> 6, m = row & 63;
    float acc = 0.f;
    if (kk < 8) {                              // p=8: m = j*8 + s
        int j = m >> 3, s = m & 7, k = kk;
        int e = j * 64 + s * 8 + k;
        for (int sp = 0; sp < SPLITS; ++sp)
            for (int qq = 0; qq < 4; ++qq)
                acc += p0[((size_t)sp * B_ + b) * 2048 + qq * 512 + e];
    } else if (kk < 24) {                      // p=16: m = j*4 + s
        int j = m >> 2, s = m & 3, k = kk - 8;
        int e = j * 64 + s * 16 + k;
        for (int sp = 0; sp < SPLITS; ++sp)
            for (int hh = 0; hh < 2; ++hh)
                acc += p1[((size_t)sp * B_ + b) * 2048 + hh * 1024 + e];
    } else if (kk < 56) {                      // p=32: m = j*2 + s
        int j = m >> 1, s = m & 1, k = kk - 24;
        int e = j * 64 + s * 32 + k;
        for (int sp = 0; sp < SPLITS; ++sp)
            acc += p2[((size_t)sp * B_ + b) * 2048 + e];
    }
    A[idx] = acc;                              // kk>=56 -> zero padding
}

// ---- weight folding: Weff[56][384] = proj_w_s @ fuse_block_s ---------------
__global__ void k_weff(const float* __restrict__ pw0, const float* __restrict__ pw1,
                       const float* __restrict__ pw2, const float* __restrict__ fuse_w,
                       float* __restrict__ Weff) {
    int idx = blockIdx.x * 256 + threadIdx.x;
    if (idx >= 56 * 384) return;
    int kk = idx / 384, df = idx % 384;
    const float* w; int sc, k;
    if (kk < 8)       { sc = 0; k = kk;      w = pw0; }
    else if (kk < 24) { sc = 1; k = kk - 8;  w = pw1; }
    else              { sc = 2; k = kk - 24; w = pw2; }
    float acc = 0.f;
    for (int d = 0; d < 128; ++d)
        acc = fmaf(w[k * 128 + d], fuse_w[(size_t)(sc * 128 + d) * 384 + df], acc);
    Weff[idx] = acc;
}

__global__ void k_beff(const float* __restrict__ pb0, const float* __restrict__ pb1,
                       const float* __restrict__ pb2, const float* __restrict__ fuse_w,
                       const float* __restrict__ fuse_b, float* __restrict__ beff) {
    int df = blockIdx.x * 256 + threadIdx.x;
    if (df >= 384) return;
    float acc = fuse_b[df];
    const float* pbs[3] = {pb0, pb1, pb2};
    for (int sc = 0; sc < 3; ++sc)
        for (int d = 0; d < 128; ++d)
            acc = fmaf(pbs[sc][d], fuse_w[(size_t)(sc * 128 + d) * 384 + df], acc);
    beff[df] = acc;
}

// ---- Wtot[64][512] = Weff @ flat_w (rows 56..63 zero) ----------------------
__global__ void k_wtot(const float* __restrict__ Weff, const float* __restrict__ flat_w,
                       float* __restrict__ Wtot) {
    int idx = blockIdx.x * 256 + threadIdx.x;  // 64*512
    int kk = idx >> 9, n = idx & 511;
    float acc = 0.f;
    if (kk < 56)
        for (int df = 0; df < 384; ++df)
            acc = fmaf(Weff[kk * 384 + df], flat_w[(size_t)df * 512 + n], acc);
    Wtot[idx] = acc;
}

__global__ void k_btot(const float* __restrict__ beff, const float* __restrict__ flat_w,
                       const float* __restrict__ flat_b, float* __restrict__ btot) {
    int n = blockIdx.x * 256 + threadIdx.x;
    if (n >= 512) return;
    float acc = flat_b[n];
    for (int df = 0; df < 384; ++df)
        acc = fmaf(beff[df], flat_w[(size_t)df * 512 + n], acc);
    btot[n] = acc;
}

// ---- final GEMM: out[1024][512] = A[1024][64] @ Wtot[64][512] + btot -------
// One 16x16 output tile per wave32; K-loop of 16 x V_WMMA_F32_16X16X4_F32.
// A operand: lanes 0-15 hold K={k0,k0+1}, lanes 16-31 hold K={k0+2,k0+3}.
// B operand mirrored (row-of-B striped across lanes); C/D: VGPR r = rows r,r+8.
__global__ void k_gemm(const float* __restrict__ A, const float* __restrict__ Wt,
                       const float* __restrict__ btot, float* __restrict__ out) {
    int wave = threadIdx.x >> 5;
    int lane = threadIdx.x & 31;
    int tile = blockIdx.x * 8 + wave;          // 0..2047
    int mt = tile >> 5, nt = tile & 31;
    int mbase = mt * 16, nbase = nt * 16;
    int half = lane >> 4, l16 = lane & 15;
    const float* Ar = A + (size_t)(mbase + l16) * 64;
    const float* Bc = Wt + nbase + l16;
    v8f acc = {};
#pragma unroll
    for (int k0 = 0; k0 < 64; k0 += 4) {
        int ka = k0 + half * 2;
        v2f av = { Ar[ka], Ar[ka + 1] };
        v2f bv = { Bc[(size_t)ka * 512], Bc[(size_t)(ka + 1) * 512] };
        acc = __builtin_amdgcn_wmma_f32_16x16x4_f32(
            false, av, false, bv, (short)0, acc, false, false);
    }
    float bias = btot[nbase + l16];
#pragma unroll
    for (int r = 0; r < 8; ++r)
        out[(size_t)(mbase + r + 8 * half) * 512 + nbase + l16] = acc[r] + bias;
}

extern "C" void kernel_launch(void* const* d_in, const int* in_sizes, int n_in,
                              void* d_out, int out_size, void* d_ws, size_t ws_size,
                              hipStream_t stream) {
    (void)in_sizes; (void)n_in; (void)out_size; (void)ws_size;
    const float* x      = (const float*)d_in[0];
    const float* pw0    = (const float*)d_in[1];
    const float* pb0    = (const float*)d_in[2];
    const float* pw1    = (const float*)d_in[3];
    const float* pb1    = (const float*)d_in[4];
    const float* pw2    = (const float*)d_in[5];
    const float* pb2    = (const float*)d_in[6];
    const float* fuse_w = (const float*)d_in[7];
    const float* fuse_b = (const float*)d_in[8];
    const float* flat_w = (const float*)d_in[9];
    const float* flat_b = (const float*)d_in[10];
    float* out = (float*)d_out;

    float* ws   = (float*)d_ws;                // ~3.5 MiB of fp32 scratch
    float* c0   = ws;                          // 512
    float* c1   = ws + 512;                    // 256
    float* c2   = ws + 768;                    // 128
    float* p0   = ws + 1024;                   // SPLITS*16*2048 = 262144
    float* p1   = p0 + SPLITS * B_ * 2048;
    float* p2   = p1 + SPLITS * B_ * 2048;
    float* A    = p2 + SPLITS * B_ * 2048;     // 1024*64
    float* Weff = A + 1024 * 64;               // 56*384
    float* beff = Weff + 56 * 384;             // 384
    float* Wtot = beff + 384;                  // 64*512
    float* btot = Wtot + 64 * 512;             // 512

    k_coef<<<1, 512, 0, stream>>>(c0, c1, c2);
    k_weff<<<(56 * 384 + 255) / 256, 256, 0, stream>>>(pw0, pw1, pw2, fuse_w, Weff);
    k_beff<<<2, 256, 0, stream>>>(pb0, pb1, pb2, fuse_w, fuse_b, beff);
    k_wtot<<<(64 * 512) / 256, 256, 0, stream>>>(Weff, flat_w, Wtot);
    k_btot<<<2, 256, 0, stream>>>(beff, flat_w, flat_b, btot);

    dim3 g1(2048 / 256, B_, SPLITS);           // one 16 MiB pass over x
    k_stage1<<<g1, 256, 0, stream>>>(x, c0, c1, c2, p0, p1, p2);
    k_stage2<<<(1024 * 64) / 256, 256, 0, stream>>>(p0, p1, p2, A);

    k_gemm<<<2048 / 8, 256, 0, stream>>>(A, Wtot, btot, out);
}